// FasterRCnn_79130477461989
// MI455X (gfx1250) — compile-verified
//
#include <hip/hip_runtime.h>

#define MAX_M 512
#define BLOCK 256

typedef int v4i __attribute__((ext_vector_type(4)));
typedef __attribute__((address_space(1))) v4i* gptr_v4i;
typedef __attribute__((address_space(3))) v4i* lptr_v4i;
typedef __attribute__((address_space(1))) int* gptr_i32;
typedef __attribute__((address_space(3))) int* lptr_i32;

__global__ __launch_bounds__(BLOCK)
void roi_assign_kernel(const float* __restrict__ props,
                       const float* __restrict__ gts,
                       const int*   __restrict__ labels,
                       float* __restrict__ out,
                       int N, int M) {
    __shared__ float4 s_box[MAX_M];
    __shared__ float  s_area[MAX_M];
    __shared__ int    s_lab[MAX_M];

    const int tid = threadIdx.x;

    // ---- Stage gt boxes + labels into LDS via CDNA5 async global->LDS DMA ----
    // Compile-time unrolled strided copy: at most MAX_M/BLOCK predicated issues,
    // no dynamic loop bookkeeping.
#pragma unroll
    for (int c = 0; c < MAX_M / BLOCK; ++c) {
        const int t = tid + c * BLOCK;
        if (t < M) {
#if __has_builtin(__builtin_amdgcn_global_load_async_to_lds_b128)
            {
                float* gnc = const_cast<float*>(gts) + 4 * t;
                __builtin_amdgcn_global_load_async_to_lds_b128(
                    (gptr_v4i)reinterpret_cast<v4i*>(gnc),
                    (lptr_v4i)reinterpret_cast<v4i*>(&s_box[t]), 0, 0);
            }
#else
            s_box[t] = ((const float4*)gts)[t];
#endif
#if __has_builtin(__builtin_amdgcn_global_load_async_to_lds_b32)
            {
                int* gnc = const_cast<int*>(labels) + t;
                __builtin_amdgcn_global_load_async_to_lds_b32(
                    (gptr_i32)gnc, (lptr_i32)&s_lab[t], 0, 0);
            }
#else
            s_lab[t] = labels[t];
#endif
        }
    }
#if __has_builtin(__builtin_amdgcn_s_wait_asynccnt)
    __builtin_amdgcn_s_wait_asynccnt(0);
#else
    asm volatile("s_wait_asynccnt 0" ::: "memory");
#endif
    __syncthreads();
#pragma unroll
    for (int c = 0; c < MAX_M / BLOCK; ++c) {
        const int t = tid + c * BLOCK;
        if (t < M) {
            float4 b = s_box[t];
            s_area[t] = (b.z - b.x) * (b.w - b.y);
        }
    }
    __syncthreads();

    const int i = blockIdx.x * BLOCK + tid;
    if (i >= N) return;

    const float4 p = ((const float4*)props)[i];
    const float pw = p.z - p.x;
    const float ph = p.w - p.y;
    const float area_a = pw * ph;

    // Two independent running-best chains (even/odd j) -> shorter serial chain.
    // Sentinel (-1, 1): first real candidate always wins (inter*1 > -1*uni).
    float bi0 = -1.0f, bu0 = 1.0f; int id0 = 0;
    float bi1 = -1.0f, bu1 = 1.0f; int id1 = 0;

#pragma unroll 8
    for (int j = 0; j + 1 < M; j += 2) {
        {
            const float4 b = s_box[j];
            const float lx = fmaxf(p.x, b.x);
            const float ly = fmaxf(p.y, b.y);
            const float rx = fminf(p.z, b.z);
            const float ry = fminf(p.w, b.w);
            const float w  = fmaxf(rx - lx, 0.0f);
            const float h  = fmaxf(ry - ly, 0.0f);
            const float inter = w * h;
            const float uni   = area_a + s_area[j] - inter;
            // inter/uni > bi0/bu0  <=>  inter*bu0 > bi0*uni  (uni, bu0 > 0)
            if (inter * bu0 > bi0 * uni) { bi0 = inter; bu0 = uni; id0 = j; }
        }
        {
            const float4 b = s_box[j + 1];
            const float lx = fmaxf(p.x, b.x);
            const float ly = fmaxf(p.y, b.y);
            const float rx = fminf(p.z, b.z);
            const float ry = fminf(p.w, b.w);
            const float w  = fmaxf(rx - lx, 0.0f);
            const float h  = fmaxf(ry - ly, 0.0f);
            const float inter = w * h;
            const float uni   = area_a + s_area[j + 1] - inter;
            if (inter * bu1 > bi1 * uni) { bi1 = inter; bu1 = uni; id1 = j + 1; }
        }
    }
    if (M & 1) {  // tail (M even in practice; kept for generality)
        const int j = M - 1;
        const float4 b = s_box[j];
        const float lx = fmaxf(p.x, b.x);
        const float ly = fmaxf(p.y, b.y);
        const float rx = fminf(p.z, b.z);
        const float ry = fminf(p.w, b.w);
        const float w  = fmaxf(rx - lx, 0.0f);
        const float h  = fmaxf(ry - ly, 0.0f);
        const float inter = w * h;
        const float uni   = area_a + s_area[j] - inter;
        if (inter * bu0 > bi0 * uni) { bi0 = inter; bu0 = uni; id0 = j; }
    }

    // Merge the two chains; on exact tie prefer the smaller index
    // (matches jnp.argmax first-occurrence semantics).
    const float c0 = bi0 * bu1;
    const float c1 = bi1 * bu0;
    const bool takeB = (c1 > c0) || ((c1 == c0) && (id1 < id0));
    const float bi  = takeB ? bi1 : bi0;
    const float bu  = takeB ? bu1 : bu0;
    const int   idx = takeB ? id1 : id0;

    // Single division per thread.
    const float max_iou = bi / bu;
    const bool pos = (max_iou >= 0.5f);

    out[i] = pos ? (float)s_lab[idx] : 0.0f;

    const float4 g = s_box[idx];
    const float gw = g.z - g.x;
    const float gh = g.w - g.y;
    const float gx = g.x + 0.5f * gw;
    const float gy = g.y + 0.5f * gh;
    const float px = p.x + 0.5f * pw;
    const float py = p.y + 0.5f * ph;

    float4 d;
    d.x = pos ? (gx - px) / pw : 0.0f;
    d.y = pos ? (gy - py) / ph : 0.0f;
    d.z = pos ? logf(gw / pw) : 0.0f;
    d.w = pos ? logf(gh / ph) : 0.0f;
    ((float4*)(out + N))[i] = d;
}

extern "C" void kernel_launch(void* const* d_in, const int* in_sizes, int n_in,
                              void* d_out, int out_size, void* d_ws, size_t ws_size,
                              hipStream_t stream) {
    const float* props  = (const float*)d_in[0];
    const float* gts    = (const float*)d_in[1];
    const int*   labels = (const int*)d_in[2];
    float* out = (float*)d_out;

    const int N = in_sizes[0] / 4;
    const int M = in_sizes[1] / 4;

    const int blocks = (N + BLOCK - 1) / BLOCK;
    roi_assign_kernel<<<blocks, BLOCK, 0, stream>>>(props, gts, labels, out, N, M);
}